// Corr1d_x_51951924412915
// MI455X (gfx1250) — compile-verified
//
#include <hip/hip_runtime.h>

// Corr1d_x on gfx1250 via V_WMMA_F32_16X16X4_F32.
// img1, img2 = [N=8, C=256, H=128, W=256] fp32 -> out = [N, 27, H, W] fp32.
// out[n,ch,h,w] = (1/C) * sum_c img1[n,c,h,w] * img2[n,c,h,w+ch-23], zero when shifted col OOB.
//
// Workgroup = 3 waves / 48 output cols. Wave m owns M-tile m and its 3 N-tiles
// (A fragment reused 3x). LDS tiles stored K-major with the K space split into
// residue classes {0,1} / {2,3} (mod 4) per column, matching the WMMA lane
// layout, so each ds_load_b128 delivers fragments for TWO K-chunks with no
// register repacking: 4 LDS loads + 6 WMMAs per 8 channels.

#define CORR_N 8
#define CORR_C 256
#define CORR_H 128
#define CORR_W 256
#define TOPC   27
#define WIN_LO (-23)       // min shift: (0-13) + (-10)
#define BLK_COLS 48        // output columns per workgroup (3 waves x 16)
#define A_COLS   48        // img1 tile columns
#define B_COLS   80        // shared img2 window columns
#define S_COL    268       // dwords per transposed column; S/4=67 odd => b128 at bank floor
#define KHALF    132       // dword offset of the k%4 in {2,3} region (33 quads => +1 quad shift)

typedef float v2f __attribute__((ext_vector_type(2)));
typedef float v4f __attribute__((ext_vector_type(4)));
typedef float v8f __attribute__((ext_vector_type(8)));

// Position of channel c within a transposed column.
__device__ __forceinline__ int kperm(int c) {
    return ((c & 2) ? KHALF : 0) + ((c >> 2) << 1) + (c & 1);
}

__global__ __launch_bounds__(96) void corr1d_wmma_kernel(
    const float* __restrict__ img1,
    const float* __restrict__ img2,
    float* __restrict__ out)
{
    const int tid  = threadIdx.x;
    const int lane = tid & 31;
    const int wave = tid >> 5;              // M-tile id 0..2
    const int wb = blockIdx.x * BLK_COLS;   // first output column of block
    const int h  = blockIdx.y;
    const int n  = blockIdx.z;

    const int HW = CORR_H * CORR_W;
    const size_t base = ((size_t)n * CORR_C * CORR_H + h) * CORR_W;

    __shared__ __align__(16) float ldsA[A_COLS * S_COL];  // ~50 KB, col-major K-permuted
    __shared__ __align__(16) float ldsB[B_COLS * S_COL];  // ~84 KB

    // Cooperative staging: global reads stay row-contiguous (coalesced),
    // LDS writes go to the transposed K-permuted layout.
    for (int e = tid; e < CORR_C * A_COLS; e += 96) {
        const int c = e / A_COLS;
        const int j = e - c * A_COLS;
        const int w = wb + j;
        float v = 0.0f;
        if ((unsigned)w < (unsigned)CORR_W)
            v = img1[base + (size_t)c * HW + w];
        ldsA[j * S_COL + kperm(c)] = v;
    }
    for (int e = tid; e < CORR_C * B_COLS; e += 96) {
        const int c = e / B_COLS;
        const int j = e - c * B_COLS;
        const int w = wb + WIN_LO + j;
        float v = 0.0f;
        if ((unsigned)w < (unsigned)CORR_W)
            v = img2[base + (size_t)c * HW + w];
        ldsB[j * S_COL + kperm(c)] = v;
    }
    __syncthreads();

    // WMMA f32 lane layout (ISA 7.12.2): lane holds col (lane&15),
    // K rows 2*(lane>>4)+{0,1} of each 4-chunk -> residue region per half-wave.
    const int i     = lane & 15;
    const int basek = (lane >> 4) ? KHALF : 0;
    const float* As = &ldsA[(wave * 16 + i) * S_COL + basek];
    const float* Bs = &ldsB[(wave * 16 + i) * S_COL + basek];

    v8f acc0[3], acc1[3];
#pragma unroll
    for (int t = 0; t < 3; ++t) {
        acc0[t] = (v8f){0.f,0.f,0.f,0.f,0.f,0.f,0.f,0.f};
        acc1[t] = (v8f){0.f,0.f,0.f,0.f,0.f,0.f,0.f,0.f};
    }

    // Per 8 channels: 1 a-b128 + 3 b-b128 + 6 WMMAs (6 independent chains).
#pragma unroll 2
    for (int k0 = 0; k0 < CORR_C; k0 += 8) {
        const int q = k0 >> 1;               // dword index within region (mult of 4)
        const v4f av = *(const v4f*)(As + q);
        const v2f a0 = av.xy;                // K-chunk k0
        const v2f a1 = av.zw;                // K-chunk k0+4
#pragma unroll
        for (int t = 0; t < 3; ++t) {
            const v4f bv = *(const v4f*)(Bs + 16 * t * S_COL + q);
            acc0[t] = __builtin_amdgcn_wmma_f32_16x16x4_f32(
                false, a0, false, bv.xy, (short)0, acc0[t], false, false);
            acc1[t] = __builtin_amdgcn_wmma_f32_16x16x4_f32(
                false, a1, false, bv.zw, (short)0, acc1[t], false, false);
        }
    }

    // Epilogue: D_t layout VGPR r -> M = r + 8*(lane>>4), Ncol = lane&15.
    // ch = 16*t + Ncol - M (valid in [0,27)); out col = wb + 16*wave + M.
    const int Ncol  = lane & 15;
    const int mbase = (lane >> 4) << 3;
    const float scale = 1.0f / (float)CORR_C;
#pragma unroll
    for (int t = 0; t < 3; ++t) {
        const v8f acc = acc0[t] + acc1[t];
#pragma unroll
        for (int r = 0; r < 8; ++r) {
            const int M    = mbase + r;
            const int ch   = 16 * t + Ncol - M;
            const int wcol = wb + 16 * wave + M;
            if ((unsigned)ch < (unsigned)TOPC && (unsigned)wcol < (unsigned)CORR_W) {
                __builtin_nontemporal_store(
                    acc[r] * scale,
                    &out[(((size_t)n * TOPC + ch) * CORR_H + h) * CORR_W + wcol]);
            }
        }
    }
}

extern "C" void kernel_launch(void* const* d_in, const int* in_sizes, int n_in,
                              void* d_out, int out_size, void* d_ws, size_t ws_size,
                              hipStream_t stream) {
    (void)in_sizes; (void)n_in; (void)out_size; (void)d_ws; (void)ws_size;
    const float* img1 = (const float*)d_in[0];
    const float* img2 = (const float*)d_in[1];
    float* out = (float*)d_out;

    dim3 grid((CORR_W + BLK_COLS - 1) / BLK_COLS, CORR_H, CORR_N);  // (6, 128, 8)
    dim3 block(96);                                                  // 3 waves
    corr1d_wmma_kernel<<<grid, block, 0, stream>>>(img1, img2, out);
}